// MinervaEnhancedLossV3_37890201485936
// MI455X (gfx1250) — compile-verified
//
#include <hip/hip_runtime.h>
#include <math.h>

#define BATCH 2048
#define NCH 10
#define HH 48
#define WW 48
#define NPIX (HH * WW)          // 2304
#define NQUAD (NPIX / 4)        // 576 float4s per channel plane
#define NTHREADS 256
#define INV_MAXDIST (1.0f / 33.94112549695428f)  // 1/sqrt(24^2+24^2)

typedef __attribute__((ext_vector_type(2))) float v2f;
typedef __attribute__((ext_vector_type(4))) float v4fx;
typedef __attribute__((ext_vector_type(8))) float v8f;
typedef __attribute__((ext_vector_type(4))) int   v4i;

// ---------------------------------------------------------------------------
// 32-lane sum via V_WMMA_F32_16X16X4_F32 (exact f32, ones-matrix row-sum trick)
//   Stage 1: A[m][k] = {x_m, 0, x_{16+m}, 0}, B = ones -> D[m][n] = x_m + x_{16+m}
//   Stages 2-5: feed pairs of D's accumulator VGPRs back as A rows
//               (rows become {r0,r1,r8,r9}, {r2,r3,r10,r11}, ...), accumulate C.
//   Result: every element of acc = sum over all 32 lanes.
// EXEC must be all-ones: call only from uniform control flow with a full block.
// ---------------------------------------------------------------------------
__device__ __forceinline__ float wmma_sum32(float x) {
  v2f ones; ones[0] = 1.0f; ones[1] = 1.0f;
  v2f a;    a[0] = x;       a[1] = 0.0f;
  v8f z = {};
  v8f d = __builtin_amdgcn_wmma_f32_16x16x4_f32(false, a, false, ones,
                                                (short)0, z, false, false);
  v8f acc = {};
  v2f p;
  p[0] = d[0]; p[1] = d[1];
  acc = __builtin_amdgcn_wmma_f32_16x16x4_f32(false, p, false, ones,
                                              (short)0, acc, false, false);
  p[0] = d[2]; p[1] = d[3];
  acc = __builtin_amdgcn_wmma_f32_16x16x4_f32(false, p, false, ones,
                                              (short)0, acc, false, false);
  p[0] = d[4]; p[1] = d[5];
  acc = __builtin_amdgcn_wmma_f32_16x16x4_f32(false, p, false, ones,
                                              (short)0, acc, false, false);
  p[0] = d[6]; p[1] = d[7];
  acc = __builtin_amdgcn_wmma_f32_16x16x4_f32(false, p, false, ones,
                                              (short)0, acc, false, false);
  return acc[0];
}

// ---------------------------------------------------------------------------
// Kernel 1: one block per batch. Streams pred once with b128 loads (each wave
// load moves 512B), computes all per-batch reductions, writes 4 partials/batch.
// ---------------------------------------------------------------------------
__global__ __launch_bounds__(NTHREADS) void minerva_per_batch_kernel(
    const float* __restrict__ pred,      // [B, C, H, W]
    const float* __restrict__ strat,     // [B, 256]
    const int*   __restrict__ targets,   // [B, H, W]
    const int*   __restrict__ inputs,    // [B, H, W]
    float* __restrict__ focalw,          // [B] sum((1-pt)^g * ce) * w_b
    float* __restrict__ ultra,           // [B] 0.85*wIoU + 0.15*strict
    float* __restrict__ copyv,           // [B] copy flag
    float* __restrict__ stratsum)        // [B] sum of 256 features
{
  const int b   = blockIdx.x;
  const int tid = threadIdx.x;

  __shared__ int   t_lds[NPIX];      // targets tile: neighbor reads hit LDS
  __shared__ float rf[4][8];
  __shared__ int   ri[3][8];
  __shared__ unsigned ru[8];

  const v4i*  tb4 = (const v4i*)(targets + (size_t)b * NPIX);
  const v4i*  ib4 = (const v4i*)(inputs  + (size_t)b * NPIX);
  const v4fx* pb4 = (const v4fx*)(pred + (size_t)b * NCH * NPIX);

  for (int q = tid; q < NQUAD; q += NTHREADS) ((v4i*)t_lds)[q] = tb4[q];
  __syncthreads();

  float facc = 0.0f, eacc = 0.0f, swacc = 0.0f;
  int strict_ok = 1, copy_ok = 1, trans = 0;
  unsigned cmask = 0u;

  for (int q = tid; q < NQUAD; q += NTHREADS) {
    // prefetch next quad-column of channel 0 (speculative; OOB is dropped)
    __builtin_prefetch(pb4 + q + NTHREADS, 0, 0);

    v4fx v4[NCH];                      // 10 channels x 4 consecutive pixels
#pragma unroll
    for (int c = 0; c < NCH; ++c) v4[c] = pb4[c * NQUAD + q];  // b128 loads
    const v4i in4 = ib4[q];

    // all 4 pixels of a quad share a row since 4 | 48
    const int h  = q / (WW / 4);
    const int w0 = (q - h * (WW / 4)) * 4;
    const float dy2 = (float)((h - HH / 2) * (h - HH / 2));
    const int p0 = q * 4;

#pragma unroll
    for (int j = 0; j < 4; ++j) {
      const int p = p0 + j;
      const int t = t_lds[p];

      // argmax (first-occurrence tie break == jnp.argmax)
      float vmax = v4[0][j];
      int amax = 0;
#pragma unroll
      for (int c = 1; c < NCH; ++c) {
        const float x = v4[c][j];
        if (x > vmax) { vmax = x; amax = c; }
      }

      // single channel pass: sum of exp AND select v[t]
      float s = 0.0f;
      float vt = v4[0][j];
#pragma unroll
      for (int c = 0; c < NCH; ++c) {
        const float x = v4[c][j];
        s += __expf(x - vmax);
        if (c == t) vt = x;
      }
      const float lse = vmax + __logf(s);

      const float ce = lse - vt;          // >= 0
      const float pt = __expf(-ce);       // in (0, 1]
      float om = 1.0f - pt;
      om = om < 0.0f ? 0.0f : om;
      // om^2.5 == om*om*sqrt(om): exact, no log/exp pair, no log(0) edge
      facc += (om * om * sqrtf(om)) * ce;

      const int w = w0 + j;
      const float dx = (float)(w - WW / 2);
      const float sw = 1.0f + 0.3f * (1.0f - sqrtf(dy2 + dx * dx) * INV_MAXDIST);
      swacc += sw;

      const int eq = (amax == t);
      eacc += eq ? sw : 0.0f;
      strict_ok &= eq;
      copy_ok   &= (amax == in4[j]);
      cmask |= 1u << t;
      if (w < WW - 1) trans += (t != t_lds[p + 1]);
      if (h < HH - 1) trans += (t != t_lds[p + WW]);
    }
  }

  // strategic features: exactly one per thread (256 per batch)
  float sv = strat[(size_t)b * 256 + tid];

  // intra-wave (wave32) reduction
#pragma unroll
  for (int off = 16; off > 0; off >>= 1) {
    facc  += __shfl_down(facc, off);
    eacc  += __shfl_down(eacc, off);
    swacc += __shfl_down(swacc, off);
    sv    += __shfl_down(sv, off);
    strict_ok &= __shfl_down(strict_ok, off);
    copy_ok   &= __shfl_down(copy_ok, off);
    trans     += __shfl_down(trans, off);
    cmask     |= __shfl_down(cmask, off);
  }

  const int wave = tid >> 5;
  const int lane = tid & 31;
  if (lane == 0) {
    rf[0][wave] = facc; rf[1][wave] = eacc; rf[2][wave] = swacc; rf[3][wave] = sv;
    ri[0][wave] = strict_ok; ri[1][wave] = copy_ok; ri[2][wave] = trans;
    ru[wave] = cmask;
  }
  __syncthreads();

  if (tid == 0) {
    float F = 0.0f, E = 0.0f, S = 0.0f, V = 0.0f;
    int st = 1, cp = 1, tr = 0;
    unsigned cm = 0u;
#pragma unroll
    for (int i = 0; i < 8; ++i) {
      F += rf[0][i]; E += rf[1][i]; S += rf[2][i]; V += rf[3][i];
      st &= ri[0][i]; cp &= ri[1][i]; tr += ri[2][i]; cm |= ru[i];
    }
    const int uniq = __popc(cm);
    const float wcoef = (uniq > 4 ? 1.3f : 1.0f) * (tr > WW ? 1.2f : 1.0f);
    focalw[b]   = F * wcoef;
    ultra[b]    = 0.85f * (E / S) + 0.15f * (st ? 1.0f : 0.0f);
    copyv[b]    = cp ? 1.0f : 0.0f;
    stratsum[b] = V;
  }
}

// ---------------------------------------------------------------------------
// Kernel 2: single block. Deterministic reduction of the 2048-length partial
// arrays; final 32-lane stage uses V_WMMA_F32_16X16X4_F32. Composes scalar.
// ---------------------------------------------------------------------------
__global__ __launch_bounds__(NTHREADS) void minerva_finalize_kernel(
    const float* __restrict__ focalw,
    const float* __restrict__ ultra,
    const float* __restrict__ copyv,
    const float* __restrict__ stratsum,
    const float* __restrict__ planning,
    const float* __restrict__ reasoning,
    float* __restrict__ out)
{
  const int tid = threadIdx.x;
  __shared__ float sd[6][NTHREADS];

  float s0 = 0.f, s1 = 0.f, s2 = 0.f, s3 = 0.f, s4 = 0.f, s5 = 0.f;
  for (int i = tid; i < BATCH; i += NTHREADS) {
    s0 += focalw[i];   s1 += ultra[i];    s2 += copyv[i];
    s3 += stratsum[i]; s4 += planning[i]; s5 += reasoning[i];
  }
  sd[0][tid] = s0; sd[1][tid] = s1; sd[2][tid] = s2;
  sd[3][tid] = s3; sd[4][tid] = s4; sd[5][tid] = s5;
  __syncthreads();

  const int lane = tid & 31;
  float tot[6];
#pragma unroll
  for (int j = 0; j < 6; ++j) {
    float x = 0.0f;
#pragma unroll
    for (int k = 0; k < NTHREADS; k += 32) x += sd[j][lane + k];
    tot[j] = wmma_sum32(x);   // uniform flow, full EXEC: WMMA legal
  }

  if (tid == 0) {
    const float invB = 1.0f / (float)BATCH;
    const float focal = tot[0] * (1.0f / ((float)BATCH * (float)NPIX));
    const float ut_mean   = tot[1] * invB;
    const float copy_mean = tot[2] * invB;

    const float transform_penalty = copy_mean * 0.5f;
    const float exact_bonus = fmaxf(-ut_mean * 5.0f, -5.0f);

    const float strat_mean = tot[3] * (1.0f / ((float)BATCH * 256.0f));
    const float creativity = (1.0f / (1.0f + __expf(-strat_mean))) * 0.1f;
    const float strategic  = (tot[4] * invB) * 0.1f;
    const float multi_step = (tot[5] * invB) * 0.1f;
    const float complexity = ut_mean * ((float)NPIX / 1225.0f) * 0.1f;

    float total = focal + transform_penalty + exact_bonus
                - creativity - strategic - multi_step - complexity;
    if (isnan(total) || isinf(total)) total = fminf(focal, 10.0f);
    out[0] = total;
  }
}

// ---------------------------------------------------------------------------
extern "C" void kernel_launch(void* const* d_in, const int* in_sizes, int n_in,
                              void* d_out, int out_size, void* d_ws, size_t ws_size,
                              hipStream_t stream) {
  (void)in_sizes; (void)n_in; (void)out_size; (void)ws_size;

  const float* pred    = (const float*)d_in[0];   // [2048,10,48,48]
  const float* strat   = (const float*)d_in[1];   // [2048,256]
  const float* plan    = (const float*)d_in[2];   // [2048]
  const float* reason  = (const float*)d_in[3];   // [2048]
  const int*   targets = (const int*)d_in[4];     // [2048,48,48]
  const int*   inputs  = (const int*)d_in[5];     // [2048,48,48]

  float* ws       = (float*)d_ws;                 // 4*2048 floats = 32 KB
  float* focalw   = ws;
  float* ultra    = ws + BATCH;
  float* copyv    = ws + 2 * BATCH;
  float* stratsum = ws + 3 * BATCH;

  minerva_per_batch_kernel<<<BATCH, NTHREADS, 0, stream>>>(
      pred, strat, targets, inputs, focalw, ultra, copyv, stratsum);

  minerva_finalize_kernel<<<1, NTHREADS, 0, stream>>>(
      focalw, ultra, copyv, stratsum, plan, reason, (float*)d_out);
}